// CustomLoss_82343112999105
// MI455X (gfx1250) — compile-verified
//
#include <hip/hip_runtime.h>
#include <hip/hip_bf16.h>
#include <stdint.h>

// Problem constants (fixed by the reference)
#define B_ROWS   1048576u
#define ROWF     114            // floats per prediction row: 2 coords + 112 logits
#define RPB      32             // rows per staged batch (== lanes, lane-per-row)
#define BATCHF   (RPB * ROWF)   // 3648 floats = 14592 bytes per batch (16B aligned)
#define NBATCH   4              // batches per wave -> 128 rows per wave
#define WPB      2              // waves per block (64 threads) -> ~57KB LDS/block
#define NBLOCKS  (B_ROWS / (RPB * NBATCH * WPB))   // 4096

// Native 1-instruction transcendentals (v_exp_f32 / v_log_f32 are base-2).
#if defined(__has_builtin)
#  if __has_builtin(__builtin_amdgcn_exp2f)
#    define EXP2F(x) __builtin_amdgcn_exp2f(x)
#  endif
#  if __has_builtin(__builtin_amdgcn_logf)
#    define LOG2F(x) __builtin_amdgcn_logf(x)
#  endif
#endif
#ifndef EXP2F
#  define EXP2F(x) exp2f(x)
#endif
#ifndef LOG2F
#  define LOG2F(x) __log2f(x)
#endif

// ---------------------------------------------------------------------------
// Stage one 32-row batch (14592B, 16B-aligned both sides) into LDS with the
// CDNA5 async copy engine: 29 x GLOBAL_LOAD_ASYNC_TO_LDS_B128 (ASYNCcnt += 29).
// INST_OFFSET is added to BOTH the global and the LDS address per the ISA, so
// a single (lds,vaddr) pair + immediate offsets covers the whole batch.
// ---------------------------------------------------------------------------
__device__ __forceinline__ void issue_batch_async(uint32_t lds_base,
                                                  const float* gbase, int lane) {
  uint32_t ldst = lds_base + (uint32_t)lane * 16u;
  const char* g = (const char*)gbase + (size_t)lane * 16u;
  #pragma unroll
  for (int j = 0; j < 28; ++j) {
    asm volatile("global_load_async_to_lds_b128 %0, %1, off offset:%2"
                 :: "v"(ldst), "v"(g), "i"(j * 512) : "memory");
  }
  if (lane < 16) {   // tail: 912 float4 = 28*32 + 16
    asm volatile("global_load_async_to_lds_b128 %0, %1, off offset:%2"
                 :: "v"(ldst), "v"(g), "i"(28 * 512) : "memory");
  }
}

__global__ __launch_bounds__(64) void loss_partial_kernel(
    const float* __restrict__ pred,   // [B_ROWS, 114]
    const float* __restrict__ tgt,    // [B_ROWS, 2]
    float* __restrict__ partial)      // [NBLOCKS]
{
  __shared__ __align__(16) float stage[WPB][2][BATCHF];
  __shared__ float wsum[WPB];

  const int tid  = threadIdx.x;
  const int wave = tid >> 5;
  const int lane = tid & 31;
  const unsigned gwave = blockIdx.x * WPB + wave;
  const unsigned row0  = gwave * (RPB * NBATCH);   // first row of this wave

  const float* buf0 = &stage[wave][0][0];
  const float* buf1 = &stage[wave][1][0];
  const uint32_t lds0 = (uint32_t)(uintptr_t)buf0;
  const uint32_t lds1 = (uint32_t)(uintptr_t)buf1;

  const float*  gbase = pred + (size_t)row0 * ROWF;
  const float2* tgt2  = (const float2*)tgt;

  // Prime the pipeline: batch 0 -> buffer 0.
  issue_batch_async(lds0, gbase, lane);

  const float L2E = 1.4426950408889634f;   // log2(e)
  const float LN2 = 0.6931471805599453f;
  float acc = 0.0f;                        // per-lane accumulator (lane-per-row)

  for (int bt = 0; bt < NBATCH; ++bt) {
    // Independent global load: overlaps the async-copy wait below.
    const float2 t = tgt2[row0 + bt * RPB + lane];  // coalesced 8B/lane

    if (bt + 1 < NBATCH) {
      // All prior LDS reads of the buffer we are about to overwrite must be
      // complete before the async engine can write it.
      asm volatile("s_wait_dscnt 0" ::: "memory");
      issue_batch_async((bt & 1) ? lds0 : lds1,
                        gbase + (size_t)(bt + 1) * BATCHF, lane);
      // 29 outstanding belong to batch bt+1; in-order completion means
      // asynccnt <= 29  ==>  batch bt has fully landed in LDS.
      asm volatile("s_wait_asynccnt 29" ::: "memory");
    } else {
      asm volatile("s_wait_asynccnt 0" ::: "memory");
    }

    const float* buf  = (bt & 1) ? buf1 : buf0;
    const float* rowp = buf + (size_t)lane * ROWF;   // this lane's row
    // dword-bank map: lane -> 114*lane mod 64 is injective over 32 lanes
    // => conflict-free; float2 reads are 8B-aligned => ds b64 loads.

    const float2 pc = *(const float2*)(rowp);        // predicted coords

    // Pass 1: max + first-occurrence argmax (strict '>' like jnp.argmax).
    // Unroll capped: bounds register pressure, keeps DS loads pipelined.
    float bv = -3.402823466e38f;
    int   bi = 0;
    #pragma unroll 8
    for (int k2 = 0; k2 < 56; ++k2) {
      const float2 x = *(const float2*)(rowp + 2 + 2 * k2);
      if (x.x > bv) { bv = x.x; bi = 2 * k2;     }
      if (x.y > bv) { bv = x.y; bi = 2 * k2 + 1; }
    }

    // Pass 2: sum of exp(x - m) via native v_exp_f32 (base-2).
    const float mL = bv * L2E;
    float s = 0.0f;
    #pragma unroll 8
    for (int k2 = 0; k2 < 56; ++k2) {
      const float2 x = *(const float2*)(rowp + 2 + 2 * k2);
      s += EXP2F(fmaf(x.x, L2E, -mL));
      s += EXP2F(fmaf(x.y, L2E, -mL));
    }

    // Target class; (v-22)/0.25 == (v-22)*4 bit-exactly (Sterbenz + pow2).
    int cx = (int)floorf((t.x - 22.0f)  * 4.0f);
    int cy = (int)floorf((t.y - 120.0f) * 4.0f);
    int tc = cx * 8 + cy;
    tc = (tc < 0) ? 0 : ((tc > 111) ? 111 : tc);   // JAX clips OOB gathers

    const float xt  = rowp[2 + tc];                 // per-lane gather from LDS
    const float ce  = fmaf(LN2, LOG2F(s), bv - xt); // -log_softmax[tc]
    const float dx  = pc.x - t.x;
    const float dy  = pc.y - t.y;
    const float mse = 0.5f * (dx * dx + dy * dy);

    acc += (bi == tc) ? mse : ce;
  }

  // One butterfly per 128 rows: reduce lane accumulators within the wave.
  #pragma unroll
  for (int off = 16; off > 0; off >>= 1) acc += __shfl_xor(acc, off, 32);
  if (lane == 0) wsum[wave] = acc;
  __syncthreads();
  if (tid == 0) {
    float sblk = 0.0f;
    #pragma unroll
    for (int w = 0; w < WPB; ++w) sblk += wsum[w];
    partial[blockIdx.x] = sblk;
  }
}

// Deterministic second stage: fixed strided per-thread sums + fixed LDS tree.
__global__ __launch_bounds__(256) void final_reduce_kernel(
    const float* __restrict__ partial, float* __restrict__ out,
    int n, float inv_b)
{
  __shared__ double sm[256];
  double s = 0.0;
  for (int i = threadIdx.x; i < n; i += 256) s += (double)partial[i];
  sm[threadIdx.x] = s;
  __syncthreads();
  #pragma unroll
  for (int stride = 128; stride > 0; stride >>= 1) {
    if (threadIdx.x < stride) sm[threadIdx.x] += sm[threadIdx.x + stride];
    __syncthreads();
  }
  if (threadIdx.x == 0) out[0] = (float)(sm[0] * (double)inv_b);
}

extern "C" void kernel_launch(void* const* d_in, const int* in_sizes, int n_in,
                              void* d_out, int out_size, void* d_ws, size_t ws_size,
                              hipStream_t stream) {
  const float* pred = (const float*)d_in[0];   // (B, 114) f32
  const float* tgt  = (const float*)d_in[1];   // (B, 2)   f32
  float* partial = (float*)d_ws;               // NBLOCKS * 4B = 16 KB scratch

  loss_partial_kernel<<<NBLOCKS, 64, 0, stream>>>(pred, tgt, partial);
  final_reduce_kernel<<<1, 256, 0, stream>>>(partial, (float*)d_out,
                                             (int)NBLOCKS, 1.0f / (float)B_ROWS);
}